// MotifDiscoveryModule_89635967467872
// MI455X (gfx1250) — compile-verified
//
#include <hip/hip_runtime.h>
#include <math.h>

// ---------------------------------------------------------------------------
// Problem constants (from reference)
// ---------------------------------------------------------------------------
#define B_    128
#define K_    16
#define D_    64
#define N_    9216          // 96*96
#define WIMG  96
#define NWAVE 8
#define NPW   (N_ / NWAVE)  // 1152 pixels per wave
#define NTILE (NPW / 16)    // 72 tiles of 16 pixels per wave
#define HPAD  68            // padded LDS row stride (floats) for h tiles
#define SPAD  21            // padded LDS row stride (floats) for pass-2 staging

#define MAPS_OFF 131072                   // B*K*D
#define ENT_OFF  (131072 + 18874368)      // + B*K*N
#define SCO_OFF  (ENT_OFF + 2048)         // + B*K

typedef float v2f __attribute__((ext_vector_type(2)));
typedef float v8f __attribute__((ext_vector_type(8)));
typedef int   v4i __attribute__((ext_vector_type(4)));

#define AS1 __attribute__((address_space(1)))
#define AS3 __attribute__((address_space(3)))

// CDNA5 async global->LDS path (guarded; sync fallback otherwise)
#if defined(__gfx1250__) && __has_builtin(__builtin_amdgcn_global_load_async_to_lds_b128) && __has_builtin(__builtin_amdgcn_s_wait_asynccnt)
#define USE_ASYNC_LDS 1
#define WAIT_ASYNC(n) __builtin_amdgcn_s_wait_asynccnt(n)
#else
#define USE_ASYNC_LDS 0
#define WAIT_ASYNC(n)
#endif

__device__ __forceinline__ v8f wmma4(v2f a, v2f b, v8f c) {
  // D = A(16x4 f32) * B(4x16 f32) + C(16x16 f32)
  return __builtin_amdgcn_wmma_f32_16x16x4_f32(false, a, false, b, (short)0, c,
                                               false, false);
}

// ---------------------------------------------------------------------------
// Prep: qkS = alpha * (mq@wq + bq) @ wk^T   [16,64],  c_q = alpha * qproj.bk
// alpha = exp(min(logit_scale, ln 10)) / (sqrt(D)*TEMP)
// ---------------------------------------------------------------------------
__global__ void motif_prep_kernel(const float* __restrict__ mq,
                                  const float* __restrict__ wq,
                                  const float* __restrict__ bq,
                                  const float* __restrict__ wk,
                                  const float* __restrict__ bk,
                                  const float* __restrict__ ls,
                                  float* __restrict__ ws) {
  __shared__ float qproj[K_ * D_];
  const int tid = threadIdx.x;
  for (int i = tid; i < K_ * D_; i += 256) {
    const int q = i >> 6, d = i & 63;
    float acc = bq[d];
    for (int e = 0; e < D_; ++e) acc += mq[q * D_ + e] * wq[e * D_ + d];
    qproj[i] = acc;
  }
  __syncthreads();
  const float alpha = expf(fminf(ls[0], 2.302585093f)) * 0.125f;
  for (int i = tid; i < K_ * D_; i += 256) {
    const int q = i >> 6, e = i & 63;
    float acc = 0.f;
    for (int d = 0; d < D_; ++d) acc += qproj[q * D_ + d] * wk[e * D_ + d];
    ws[i] = alpha * acc;
  }
  if (tid < K_) {
    float acc = 0.f;
    for (int d = 0; d < D_; ++d) acc += qproj[tid * D_ + d] * bk[d];
    ws[K_ * D_ + tid] = alpha * acc;
  }
}

// ---------------------------------------------------------------------------
// Main fused kernel: 1 workgroup (8 wave32) per batch element
// ---------------------------------------------------------------------------
__global__ void __launch_bounds__(256) motif_main_kernel(
    const float* __restrict__ h, const float* __restrict__ wv,
    const float* __restrict__ bv, const float* __restrict__ ln_g,
    const float* __restrict__ ln_b, const float* __restrict__ sw1,
    const float* __restrict__ sb1, const float* __restrict__ sw2,
    const float* __restrict__ sb2, const float* __restrict__ ws,
    float* __restrict__ out) {
  __shared__ float hbuf[NWAVE][2][16 * HPAD];   // double-buffered h tiles
  __shared__ float aggLDS[K_ * D_];             // attn-weighted h accumulator
  __shared__ float statLDS[NWAVE][K_][4];       // per-wave (m, s, t)
  __shared__ float embLDS[K_ * D_];
  __shared__ float hscLDS[K_ * D_];
  __shared__ float mvLDS[K_][2];                // LN mean / rstd
  __shared__ float csLDS[K_][2];                // combined (s, z)

  const int tid = threadIdx.x;
  const int w = tid >> 5;
  const int lane = tid & 31;
  const int lq = lane & 15;   // query index in B/C fragments; pixel row in A
  const int hi = lane >> 4;
  const int b = blockIdx.x;
  const float* __restrict__ hb = h + (size_t)b * N_ * D_;

  for (int i = tid; i < K_ * D_; i += 256) aggLDS[i] = 0.f;

  // Persistent B fragments of qkS^T: lane lq holds qkS[lq][4j+2hi .. +1]
  v2f bqf[16];
#pragma unroll
  for (int j = 0; j < 16; ++j) {
    const float* p = ws + lq * D_ + 4 * j + 2 * hi;
    bqf[j][0] = p[0];
    bqf[j][1] = p[1];
  }
  const float cq = ws[K_ * D_ + lq];

  // Gaussian bias constants: 4x4 grid of query centers in [0,1]
  const float cx = (float)(lq & 3) * (1.0f / 3.0f);
  const float cy = (float)(lq >> 2) * (1.0f / 3.0f);

  float* const bufA = &hbuf[w][0][0];
  float* const bufB = &hbuf[w][1][0];
  const int nstart = w * NPW;

  // ---- issue one 16x64 h tile (4KB) global -> LDS ----
  auto issue_tile = [&](float* dst, int n0) {
#pragma unroll
    for (int i = 0; i < 8; ++i) {
      const int idx = i * 32 + lane;
      const int row = idx >> 4, col4 = idx & 15;
      const float* g = hb + (size_t)(n0 + row) * D_ + col4 * 4;
      float* l = dst + row * HPAD + col4 * 4;
#if USE_ASYNC_LDS
      __builtin_amdgcn_global_load_async_to_lds_b128((AS1 v4i*)g, (AS3 v4i*)l,
                                                     0, 0);
#else
      *(float4*)l = *(const float4*)g;
#endif
    }
  };

  v8f cagg[4];
#pragma unroll
  for (int dc = 0; dc < 4; ++dc) cagg[dc] = (v8f){};

  float m = -1e30f, s = 0.f, tac = 0.f;

  // ========================== PASS 1 ==========================
  issue_tile(bufA, nstart);
  for (int t = 0; t < NTILE; ++t) {
    const int n0 = nstart + t * 16;
    float* hl = (t & 1) ? bufB : bufA;
    if (t + 1 < NTILE) {
      issue_tile((t & 1) ? bufA : bufB, n0 + 16);
      WAIT_ASYNC(8);
    } else {
      WAIT_ASYNC(0);
    }

    // logits tile C'[pixel=r+8hi (regs), query=lq (lanes)]
    v8f cl = {};
#pragma unroll
    for (int j = 0; j < 16; ++j) {
      v2f a = *(const v2f*)(hl + lq * HPAD + 4 * j + 2 * hi);
      cl = wmma4(a, bqf[j], cl);
    }

    // bias + tile max (16 | 96 -> the whole tile sits in one image row)
    const int y = n0 / WIMG;
    const int x0 = n0 - y * WIMG;
    const float py = (float)y * (1.0f / 95.0f);
    const float dy2 = (cy - py) * (cy - py);
    float l[8];
    float tmax = -1e30f;
#pragma unroll
    for (int r = 0; r < 8; ++r) {
      const float px = (float)(x0 + r + 8 * hi) * (1.0f / 95.0f);
      const float dx = cx - px;
      const float lv = cl[r] + cq - 8.0f * (dx * dx + dy2);
      l[r] = lv;
      tmax = fmaxf(tmax, lv);
    }
    tmax = fmaxf(tmax, __shfl_xor(tmax, 16));

    // online softmax update (stats per query, duplicated in both halves)
    const float mn = fmaxf(m, tmax);
    const float corr = expf(m - mn);
    float p[8];
    float ps = 0.f, pt = 0.f;
#pragma unroll
    for (int r = 0; r < 8; ++r) {
      const float pv = expf(l[r] - mn);
      p[r] = pv;
      ps += pv;
      pt += pv * l[r];
    }
    ps += __shfl_xor(ps, 16);
    pt += __shfl_xor(pt, 16);
    s = s * corr + ps;
    tac = tac * corr + pt;
    m = mn;

    // rescale aggH accumulators: row q = r + 8hi needs corr of query q
#pragma unroll
    for (int r = 0; r < 8; ++r) {
      const float f = __shfl(corr, r + 8 * hi);
#pragma unroll
      for (int dc = 0; dc < 4; ++dc) cagg[dc][r] *= f;
    }

    // aggH[q, d] += P^T @ h : A = P fragments built via cross-half shuffles
#pragma unroll
    for (int c = 0; c < 4; ++c) {
      const int r0 = (4 * c) & 7;
      const float send0 = hi ? p[r0] : p[r0 + 2];
      const float send1 = hi ? p[r0 + 1] : p[r0 + 3];
      const float got0 = __shfl_xor(send0, 16);
      const float got1 = __shfl_xor(send1, 16);
      const float loc0 = hi ? p[r0 + 2] : p[r0];
      const float loc1 = hi ? p[r0 + 3] : p[r0 + 1];
      const bool rem = (c < 2) ? (hi != 0) : (hi == 0);
      v2f af;
      af[0] = rem ? got0 : loc0;
      af[1] = rem ? got1 : loc1;
      const float* hr = hl + (4 * c + 2 * hi) * HPAD;
#pragma unroll
      for (int dc = 0; dc < 4; ++dc) {
        v2f bf;
        bf[0] = hr[dc * 16 + lq];
        bf[1] = hr[HPAD + dc * 16 + lq];
        cagg[dc] = wmma4(af, bf, cagg[dc]);
      }
    }
  }

  // ================== cross-wave softmax combine ==================
  if (lane < 16) {
    statLDS[w][lq][0] = m;
    statLDS[w][lq][1] = s;
    statLDS[w][lq][2] = tac;
  }
  __syncthreads();
  float mc = -1e30f;
#pragma unroll
  for (int ww = 0; ww < NWAVE; ++ww) mc = fmaxf(mc, statLDS[ww][lq][0]);
  float sc = 0.f, tc = 0.f;
#pragma unroll
  for (int ww = 0; ww < NWAVE; ++ww) {
    const float e = expf(statLDS[ww][lq][0] - mc);
    sc += statLDS[ww][lq][1] * e;
    tc += statLDS[ww][lq][2] * e;
  }
  const float zq = mc + logf(sc);   // attn = exp(l - zq)

  // fold this wave's aggH into shared accumulator (scaled to global max)
  const float g = expf(m - mc);
#pragma unroll
  for (int r = 0; r < 8; ++r) {
    const float f = __shfl(g, r + 8 * hi);
    const int q = r + 8 * hi;
#pragma unroll
    for (int dc = 0; dc < 4; ++dc)
      atomicAdd(&aggLDS[q * D_ + dc * 16 + lq], cagg[dc][r] * f);
  }
  if (w == 0 && lane < 16) {
    csLDS[lq][0] = sc;
    csLDS[lq][1] = zq;
    out[ENT_OFF + b * K_ + lq] = zq - tc / sc;  // entropy = m + ln s - E[l]
  }
  __syncthreads();

  // ======================= epilogue (tiny) =======================
  for (int i = tid; i < K_ * D_; i += 256) aggLDS[i] *= (1.0f / csLDS[i >> 6][0]);
  __syncthreads();
  for (int i = tid; i < K_ * D_; i += 256) {      // emb = aggN @ wv + bv
    const int q = i >> 6, d = i & 63;
    float acc = bv[d];
    for (int e = 0; e < D_; ++e) acc += aggLDS[q * D_ + e] * wv[e * D_ + d];
    embLDS[i] = acc;
  }
  __syncthreads();
  if (tid < K_) {                                 // LayerNorm stats
    float mu = 0.f;
    for (int d = 0; d < D_; ++d) mu += embLDS[tid * D_ + d];
    mu *= (1.0f / D_);
    float va = 0.f;
    for (int d = 0; d < D_; ++d) {
      const float dd = embLDS[tid * D_ + d] - mu;
      va += dd * dd;
    }
    mvLDS[tid][0] = mu;
    mvLDS[tid][1] = rsqrtf(va * (1.0f / D_) + 1e-5f);
  }
  __syncthreads();
  for (int i = tid; i < K_ * D_; i += 256) {
    const int q = i >> 6, d = i & 63;
    const float e = ln_g[d] * (embLDS[i] - mvLDS[q][0]) * mvLDS[q][1] + ln_b[d];
    embLDS[i] = e;
    out[(size_t)b * K_ * D_ + i] = e;
  }
  __syncthreads();
  for (int i = tid; i < K_ * D_; i += 256) {      // score head: GELU(emb@sw1+sb1)
    const int q = i >> 6, d = i & 63;
    float acc = sb1[d];
    for (int e = 0; e < D_; ++e) acc += embLDS[q * D_ + e] * sw1[e * D_ + d];
    hscLDS[i] = 0.5f * acc * (1.0f + erff(acc * 0.70710678118654752f));
  }
  __syncthreads();
  if (tid < K_) {
    float acc = sb2[0];
    for (int d = 0; d < D_; ++d) acc += hscLDS[tid * D_ + d] * sw2[d];
    out[SCO_OFF + b * K_ + tid] = acc;
  }

  // ========================== PASS 2 ==========================
  // Recompute logits (h is L2-resident) and emit attn maps through an LDS
  // transpose stage so global stores are contiguous per query row.
  float* const stage = &hbuf[w][0][0];            // 64 x SPAD floats
  float* const mapsb = out + MAPS_OFF + (size_t)b * K_ * N_;
  for (int gi = 0; gi < NTILE / 4; ++gi) {
    const int ng = nstart + gi * 64;
    if (gi + 1 < NTILE / 4)
      __builtin_prefetch(hb + (size_t)(ng + 64 + lane) * D_, 0, 1);
#pragma unroll
    for (int tt = 0; tt < 4; ++tt) {
      const int n0 = ng + tt * 16;
      const float* ga = hb + (size_t)(n0 + lq) * D_ + 2 * hi;
      v8f cl = {};
#pragma unroll
      for (int j = 0; j < 16; ++j) {
        v2f a = *(const v2f*)(ga + 4 * j);
        cl = wmma4(a, bqf[j], cl);
      }
      const int y = n0 / WIMG;
      const int x0 = n0 - y * WIMG;
      const float py = (float)y * (1.0f / 95.0f);
      const float dy2 = (cy - py) * (cy - py);
#pragma unroll
      for (int r = 0; r < 8; ++r) {
        const float px = (float)(x0 + r + 8 * hi) * (1.0f / 95.0f);
        const float dx = cx - px;
        const float lv = cl[r] + cq - 8.0f * (dx * dx + dy2);
        stage[(tt * 16 + r + 8 * hi) * SPAD + lq] = expf(lv - zq);
      }
    }
    // flush: lane pair (lq, lq+16) writes query row lq, 128B each
#pragma unroll
    for (int k = 0; k < 8; ++k) {
      const int nl = 32 * hi + 4 * k;
      float4 vv;
      vv.x = stage[(nl + 0) * SPAD + lq];
      vv.y = stage[(nl + 1) * SPAD + lq];
      vv.z = stage[(nl + 2) * SPAD + lq];
      vv.w = stage[(nl + 3) * SPAD + lq];
      *(float4*)(mapsb + (size_t)lq * N_ + ng + nl) = vv;
    }
  }
}

// ---------------------------------------------------------------------------
extern "C" void kernel_launch(void* const* d_in, const int* in_sizes, int n_in,
                              void* d_out, int out_size, void* d_ws,
                              size_t ws_size, hipStream_t stream) {
  const float* h_pixel = (const float*)d_in[0];
  const float* mq = (const float*)d_in[1];
  const float* wq = (const float*)d_in[2];
  const float* bq = (const float*)d_in[3];
  const float* wk = (const float*)d_in[4];
  const float* bk = (const float*)d_in[5];
  const float* wv = (const float*)d_in[6];
  const float* bv = (const float*)d_in[7];
  const float* ln_g = (const float*)d_in[8];
  const float* ln_b = (const float*)d_in[9];
  const float* sw1 = (const float*)d_in[10];
  const float* sb1 = (const float*)d_in[11];
  const float* sw2 = (const float*)d_in[12];
  const float* sb2 = (const float*)d_in[13];
  const float* ls = (const float*)d_in[14];
  float* out = (float*)d_out;
  float* ws = (float*)d_ws;

  motif_prep_kernel<<<1, 256, 0, stream>>>(mq, wq, bq, wk, bk, ls, ws);
  motif_main_kernel<<<B_, 256, 0, stream>>>(h_pixel, wv, bv, ln_g, ln_b, sw1,
                                            sb1, sw2, sb2, ws, out);
}